// TemporalGCN_61701500174742
// MI455X (gfx1250) — compile-verified
//
#include <hip/hip_runtime.h>
#include <math.h>

// ---------------- problem constants (match reference) ----------------
#define FEAT    128     // F_IN == H
#define G_BATCH 64
#define LSTMH   256
#define OUT_DIM 32
#define LAYERS  3
#define WTS     130     // padded LDS row stride for transposed W (bank-friendly, keeps 8B align for even k)

typedef __attribute__((ext_vector_type(2))) float v2f;
typedef __attribute__((ext_vector_type(8))) float v8f;

// ---------------- utility ----------------
__global__ void zero_f32(float* __restrict__ p, long long n) {
  long long i = (long long)blockIdx.x * blockDim.x + threadIdx.x;
  long long stride = (long long)gridDim.x * blockDim.x;
  for (; i < n; i += stride) p[i] = 0.0f;
}

// deg[col[e]] += 1 (self-loop "+1" folded into dinv_kernel)
__global__ void degree_kernel(const int* __restrict__ col, float* __restrict__ deg, int E) {
  int e = blockIdx.x * blockDim.x + threadIdx.x;
  if (e < E) unsafeAtomicAdd(&deg[col[e]], 1.0f);
}

__global__ void dinv_kernel(float* __restrict__ d, int n) {
  int i = blockIdx.x * blockDim.x + threadIdx.x;
  if (i < n) d[i] = rsqrtf(d[i] + 1.0f);  // +1 = self loop; always > 0
}

// ---------------- t = in[N,128] @ W[128,128], fp32 WMMA ----------------
// One block = 256 threads = 8 waves; each wave owns a 16-row slab and all 128 cols.
// W is staged transposed in LDS (Wt[n][k]) so B-fragments load as aligned float2.
__global__ __launch_bounds__(256)
void gemm128_wmma(const float* __restrict__ in, const float* __restrict__ W,
                  float* __restrict__ out, int nrows) {
  __shared__ float Wt[FEAT * WTS];
  for (int i = threadIdx.x; i < FEAT * FEAT; i += 256) {
    int k = i >> 7, n = i & 127;
    Wt[n * WTS + k] = W[i];           // transpose: Wt[n][k] = W[k][n]
  }
  __syncthreads();

  const int lane  = threadIdx.x & 31;
  const int wave  = threadIdx.x >> 5;
  const int m     = lane & 15;        // A: lane -> row M; C/D: lane -> col N
  const int khalf = lane >> 4;        // A/B: which K-pair this half-wave holds
  const int rowBase = blockIdx.x * 128 + wave * 16;
  if (rowBase >= nrows) return;       // N % 16 == 0, waves fully in or out

  v8f acc[8];
  #pragma unroll
  for (int nt = 0; nt < 8; ++nt) acc[nt] = (v8f){0.f,0.f,0.f,0.f,0.f,0.f,0.f,0.f};

  const float* arow = in + (size_t)(rowBase + m) * FEAT;
  #pragma unroll 4
  for (int kk = 0; kk < 32; ++kk) {                 // K = 128 in steps of 4
    const int k0 = kk * 4 + khalf * 2;              // even -> 8B aligned
    v2f a = *(const v2f*)(arow + k0);               // A frag: lanes 0-15 K=0,1; 16-31 K=2,3
    #pragma unroll
    for (int nt = 0; nt < 8; ++nt) {
      v2f b = *(const v2f*)(&Wt[(nt * 16 + m) * WTS + k0]);   // B frag: B[k][n], K-pair per half-wave
      acc[nt] = __builtin_amdgcn_wmma_f32_16x16x4_f32(
          /*neg_a=*/false, a, /*neg_b=*/false, b,
          /*c_mod=*/(short)0, acc[nt], /*reuse_a=*/false, /*reuse_b=*/false);
    }
  }

  // D layout: VGPR r -> (M=r, N=lane) for lanes 0-15, (M=r+8, N=lane-16) for 16-31
  const int n = lane & 15;
  const int mOff = khalf * 8;
  #pragma unroll
  for (int nt = 0; nt < 8; ++nt)
    #pragma unroll
    for (int r = 0; r < 8; ++r)
      out[(size_t)(rowBase + mOff + r) * FEAT + nt * 16 + n] = acc[nt][r];
}

// ---------------- edge scatter: agg[dst] += t[src] * dinv[src]*dinv[dst] ----------------
// One wave per edge, float4 per lane (32*4 = 128 feats). L2-resident atomics.
__global__ __launch_bounds__(256)
void edge_scatter(const int* __restrict__ ei, const float* __restrict__ t,
                  const float* __restrict__ dinv, float* __restrict__ agg, int E) {
  const int lane = threadIdx.x & 31;
  const int e = blockIdx.x * (blockDim.x >> 5) + (threadIdx.x >> 5);
  if (e >= E) return;
  const int src = ei[e];
  const int dst = ei[(size_t)E + e];
  const float nrm = dinv[src] * dinv[dst];
  const float4 v = *(const float4*)(t + (size_t)src * FEAT + lane * 4);
  float* p = agg + (size_t)dst * FEAT + lane * 4;
  unsafeAtomicAdd(p + 0, v.x * nrm);
  unsafeAtomicAdd(p + 1, v.y * nrm);
  unsafeAtomicAdd(p + 2, v.z * nrm);
  unsafeAtomicAdd(p + 3, v.w * nrm);
}

// h = relu(agg + t * dinv^2 (self-loop) + bias), in place on agg
__global__ void finalize_kernel(float* __restrict__ agg, const float* __restrict__ t,
                                const float* __restrict__ dinv, const float* __restrict__ bias,
                                long long total) {
  long long idx = (long long)blockIdx.x * blockDim.x + threadIdx.x;
  long long stride = (long long)gridDim.x * blockDim.x;
  for (; idx < total; idx += stride) {
    const int i = (int)(idx >> 7), f = (int)(idx & 127);
    const float d = dinv[i];
    agg[idx] = fmaxf(agg[idx] + t[idx] * d * d + bias[f], 0.0f);
  }
}

// ---------------- global mean pool ----------------
__global__ __launch_bounds__(256)
void pool_scatter(const float* __restrict__ h, const int* __restrict__ batch,
                  float* __restrict__ pooled, float* __restrict__ cnt, int n) {
  const int lane = threadIdx.x & 31;
  const int node = blockIdx.x * (blockDim.x >> 5) + (threadIdx.x >> 5);
  if (node >= n) return;
  const int g = batch[node];
  const float4 v = *(const float4*)(h + (size_t)node * FEAT + lane * 4);
  float* p = pooled + (size_t)g * FEAT + lane * 4;
  unsafeAtomicAdd(p + 0, v.x);
  unsafeAtomicAdd(p + 1, v.y);
  unsafeAtomicAdd(p + 2, v.z);
  unsafeAtomicAdd(p + 3, v.w);
  if (lane == 0) unsafeAtomicAdd(&cnt[g], 1.0f);
}

__global__ void pool_div(float* __restrict__ pooled, const float* __restrict__ cnt) {
  int i = blockIdx.x * blockDim.x + threadIdx.x;
  if (i < G_BATCH * FEAT) pooled[i] /= fmaxf(cnt[i >> 7], 1.0f);
}

// ---------------- single-step LSTM (h0 = c0 = 0 => W_hh term and f-gate drop) ----------
__global__ void lstm_kernel(const float* __restrict__ pooled, const float* __restrict__ W_ih,
                            const float* __restrict__ b_ih, const float* __restrict__ b_hh,
                            float* __restrict__ hn) {
  int j = blockIdx.x * blockDim.x + threadIdx.x;       // 0 .. G*LSTMH-1
  if (j >= G_BATCH * LSTMH) return;
  const int g = j >> 8, u = j & 255;
  const float* p  = pooled + (size_t)g * FEAT;
  const float* wi = W_ih + (size_t)(0 * LSTMH + u) * FEAT;   // gate i
  const float* wg = W_ih + (size_t)(2 * LSTMH + u) * FEAT;   // gate g
  const float* wo = W_ih + (size_t)(3 * LSTMH + u) * FEAT;   // gate o
  float si = b_ih[u] + b_hh[u];
  float sg = b_ih[2 * LSTMH + u] + b_hh[2 * LSTMH + u];
  float so = b_ih[3 * LSTMH + u] + b_hh[3 * LSTMH + u];
  for (int k = 0; k < FEAT; ++k) {
    const float x = p[k];
    si += x * wi[k]; sg += x * wg[k]; so += x * wo[k];
  }
  const float c = (1.0f / (1.0f + expf(-si))) * tanhf(sg);
  hn[j] = (1.0f / (1.0f + expf(-so))) * tanhf(c);
}

__global__ void out_kernel(const float* __restrict__ hn, const float* __restrict__ W_out,
                           const float* __restrict__ b_out, float* __restrict__ out) {
  int idx = blockIdx.x * blockDim.x + threadIdx.x;     // 0 .. G*OUT-1
  if (idx >= G_BATCH * OUT_DIM) return;
  const int g = idx >> 5, o = idx & 31;
  const float* h = hn + (size_t)g * LSTMH;
  const float* w = W_out + (size_t)o * LSTMH;
  float s = b_out[o];
  for (int k = 0; k < LSTMH; ++k) s += h[k] * w[k];
  out[idx] = s;
}

// ---------------- launcher ----------------
extern "C" void kernel_launch(void* const* d_in, const int* in_sizes, int n_in,
                              void* d_out, int out_size, void* d_ws, size_t ws_size,
                              hipStream_t stream) {
  const float* x     = (const float*)d_in[0];
  const int*   ei    = (const int*)  d_in[1];
  const int*   batch = (const int*)  d_in[2];
  const float* Wg    = (const float*)d_in[3];
  const float* bg    = (const float*)d_in[4];
  const float* W_ih  = (const float*)d_in[5];
  /* d_in[6] = W_hh unused: h0 == 0 */
  const float* b_ih  = (const float*)d_in[7];
  const float* b_hh  = (const float*)d_in[8];
  const float* W_out = (const float*)d_in[9];
  const float* b_out = (const float*)d_in[10];
  float* out = (float*)d_out;

  const int N = in_sizes[0] / FEAT;     // 50000
  const int E = in_sizes[1] / 2;        // 1600000
  const long long NF = (long long)N * FEAT;

  // workspace layout (~77 MB)
  float* T      = (float*)d_ws;         // [N,128] transformed features
  float* A0     = T + NF;               // [N,128] agg / h ping
  float* A1     = A0 + NF;              // [N,128] agg / h pong
  float* dinv   = A1 + NF;              // [N]
  float* pooled = dinv + N;             // [G,128]
  float* cnt    = pooled + G_BATCH * FEAT;  // [G]
  float* hn     = cnt + G_BATCH;        // [G,256]

  const int B = 256;
  const int gridNF   = (int)((NF + B - 1) / B) > 4096 ? 4096 : (int)((NF + B - 1) / B);
  const int gridE    = (E + (B / 32) - 1) / (B / 32);        // 1 wave per edge
  const int gridNode = (N + (B / 32) - 1) / (B / 32);        // 1 wave per node
  const int gridGemm = (N + 127) / 128;

  // 1) degree -> dinv  (dinv buffer doubles as deg accumulator)
  zero_f32<<<(N + B - 1) / B, B, 0, stream>>>(dinv, N);
  degree_kernel<<<(E + B - 1) / B, B, 0, stream>>>(ei + E, dinv, E);
  dinv_kernel<<<(N + B - 1) / B, B, 0, stream>>>(dinv, N);

  // 2) three GCN layers: in -> T (WMMA gemm) -> edge scatter into agg -> fused self-loop+bias+relu
  const float* lin[LAYERS]  = { x,  A0, A1 };
  float*       lagg[LAYERS] = { A0, A1, A0 };
  for (int l = 0; l < LAYERS; ++l) {
    gemm128_wmma<<<gridGemm, B, 0, stream>>>(lin[l], Wg + (size_t)l * FEAT * FEAT, T, N);
    zero_f32<<<gridNF, B, 0, stream>>>(lagg[l], NF);
    edge_scatter<<<gridE, B, 0, stream>>>(ei, T, dinv, lagg[l], E);
    finalize_kernel<<<gridNF, B, 0, stream>>>(lagg[l], T, dinv, bg + (size_t)l * FEAT, NF);
  }

  // 3) global mean pool (pooled + cnt are contiguous -> one zero pass)
  zero_f32<<<(G_BATCH * FEAT + G_BATCH + B - 1) / B, B, 0, stream>>>(pooled, G_BATCH * FEAT + G_BATCH);
  pool_scatter<<<gridNode, B, 0, stream>>>(lagg[LAYERS - 1], batch, pooled, cnt, N);
  pool_div<<<(G_BATCH * FEAT + B - 1) / B, B, 0, stream>>>(pooled, cnt);

  // 4) LSTM step + output projection
  lstm_kernel<<<(G_BATCH * LSTMH + B - 1) / B, B, 0, stream>>>(pooled, W_ih, b_ih, b_hh, hn);
  out_kernel<<<(G_BATCH * OUT_DIM + B - 1) / B, B, 0, stream>>>(hn, W_out, b_out, out);
}